// MeshReadoutNetwork_68547678044331
// MI455X (gfx1250) — compile-verified
//
#include <hip/hip_runtime.h>

typedef __attribute__((ext_vector_type(2))) float v2f;
typedef __attribute__((ext_vector_type(8))) float v8f;

#define HID 64

__device__ __forceinline__ void atomAddF(float* p, float v) {
    unsafeAtomicAdd(p, v);   // lowers to global_atomic_add_f32 (L2 atomic)
}

// ---------------- zero ----------------
__global__ void k_zero(float* __restrict__ p, int n) {
    int i = blockIdx.x * blockDim.x + threadIdx.x;
    if (i < n) p[i] = 0.0f;
}

// ---------------- degree count ----------------
__global__ void k_degree(const int* __restrict__ src, const int* __restrict__ dst,
                         float* __restrict__ deg_out, float* __restrict__ deg_in, int e) {
    int i = blockIdx.x * blockDim.x + threadIdx.x;
    if (i < e) {
        atomAddF(&deg_out[src[i]], 1.0f);
        atomAddF(&deg_in[dst[i]], 1.0f);
    }
}

// ---------------- deg -> rsqrt norm (in place, covers both arrays) ----------------
__global__ void k_norm(float* __restrict__ d, int n) {
    int i = blockIdx.x * blockDim.x + threadIdx.x;
    if (i < n) {
        float x = d[i];
        d[i] = (x > 0.0f) ? rsqrtf(x) : 0.0f;
    }
}

// ---------------- h[i,c] = feat[i,c] * norm_src[i]  (d = 1<<shift) ----------------
__global__ void k_scale(const float* __restrict__ f, const float* __restrict__ ns,
                        float* __restrict__ h, int n, int shift) {
    int i = blockIdx.x * blockDim.x + threadIdx.x;
    int total = n << shift;
    if (i < total) h[i] = f[i] * ns[i >> shift];
}

// ---------------- scatter-add: agg[dst[e], :] += h[src[e], :] ----------------
// one thread per (edge, 4-float chunk); qshift = log2(d/4)
__global__ void k_scatter(const int* __restrict__ src, const int* __restrict__ dst,
                          const float* __restrict__ h, float* __restrict__ agg,
                          int e, int qshift, int d) {
    int i = blockIdx.x * blockDim.x + threadIdx.x;
    int total = e << qshift;
    if (i >= total) return;
    int ei = i >> qshift;
    int q  = i & ((1 << qshift) - 1);
    int s = src[ei];
    int t = dst[ei];
    const float4* hv = (const float4*)(h + (size_t)s * d);
    float4 v = hv[q];
    float* o = agg + (size_t)t * d + q * 4;
    atomAddF(o + 0, v.x);
    atomAddF(o + 1, v.y);
    atomAddF(o + 2, v.z);
    atomAddF(o + 3, v.w);
}

// ---------------- out = act( diag(rowscale) * A[rows,K] @ W[K,NOUT] ) ----------------
// One wave (32 lanes) per 16x16 output tile; V_WMMA_F32_16X16X4_F32 over K in steps of 4.
// A fragment (ISA 7.12.2, 32-bit 16x4): lanes 0-15 M=lane hold K={k0,k0+1}; lanes 16-31 hold K={k0+2,k0+3}.
// B fragment (4x16 dual striping):      lanes 0-15 N=lane hold K={k0,k0+1}; lanes 16-31 hold K={k0+2,k0+3}.
// C/D: VGPR r, lanes 0-15 -> (M=r, N=lane); lanes 16-31 -> (M=r+8, N=lane-16).
__global__ void k_gemm_wmma(const float* __restrict__ A, const float* __restrict__ rowscale,
                            const float* __restrict__ W, float* __restrict__ out,
                            int K, int NOUT, int relu) {
    int lane = threadIdx.x;
    int half = lane >> 4;
    int lr   = lane & 15;
    int rowA = blockIdx.x * 16 + lr;
    int colB = blockIdx.y * 16 + lr;
    float sc = rowscale[rowA];
    const float* arow = A + (size_t)rowA * K;

    v8f acc = {};
    for (int k0 = 0; k0 < K; k0 += 4) {
        int ka = k0 + (half << 1);
        v2f a, b;
        a.x = arow[ka] * sc;
        a.y = arow[ka + 1] * sc;
        b.x = W[(size_t)ka * NOUT + colB];
        b.y = W[(size_t)(ka + 1) * NOUT + colB];
        acc = __builtin_amdgcn_wmma_f32_16x16x4_f32(
            /*neg_a=*/false, a, /*neg_b=*/false, b,
            /*c_mod=*/(short)0, acc, /*reuse_a=*/false, /*reuse_b=*/false);
    }

    int rbase = blockIdx.x * 16 + (half << 3);
#pragma unroll
    for (int r = 0; r < 8; ++r) {
        float v = acc[r];
        if (relu) v = fmaxf(v, 0.0f);
        out[(size_t)(rbase + r) * NOUT + colB] = v;
    }
}

// ---------------- segment sum over sorted graph_ids ----------------
// block = 64 threads (one per hidden column); block walks a contiguous node range,
// accumulating in registers, flushing atomics only at graph-id transitions.
__global__ void k_graphsum(const float* __restrict__ h, const int* __restrict__ gid,
                           float* __restrict__ sums, float* __restrict__ counts,
                           int n, int chunk) {
    int c = threadIdx.x;                 // 0..63
    int start = blockIdx.x * chunk;
    int end = min(n, start + chunk);
    if (start >= end) return;
    int cur = gid[start];
    float acc = 0.0f, cnt = 0.0f;
    for (int i = start; i < end; ++i) {
        int g = gid[i];                  // uniform across block
        if (g != cur) {
            atomAddF(&sums[(size_t)cur * HID + c], acc);
            if (c == 0) atomAddF(&counts[cur], cnt);
            acc = 0.0f; cnt = 0.0f; cur = g;
        }
        acc += h[(size_t)i * HID + c];
        cnt += 1.0f;
    }
    atomAddF(&sums[(size_t)cur * HID + c], acc);
    if (c == 0) atomAddF(&counts[cur], cnt);
}

__global__ void k_rscale(const float* __restrict__ counts, float* __restrict__ rs) {
    int i = threadIdx.x;
    rs[i] = 1.0f / fmaxf(counts[i], 1.0f);
}

extern "C" void kernel_launch(void* const* d_in, const int* in_sizes, int n_in,
                              void* d_out, int out_size, void* d_ws, size_t ws_size,
                              hipStream_t stream) {
    const float* feats = (const float*)d_in[0];
    const int*   src   = (const int*)d_in[1];
    const int*   dst   = (const int*)d_in[2];
    const int*   gid   = (const int*)d_in[3];
    const float* W1    = (const float*)d_in[4];
    const float* W2    = (const float*)d_in[5];
    const float* Wc    = (const float*)d_in[6];
    float*       out   = (float*)d_out;

    const int N = in_sizes[3];   // 50000 (multiple of 16)
    const int E = in_sizes[1];   // 800000
    const int G = 64;

    float* ws       = (float*)d_ws;
    float* norm_src = ws;                            // N   (deg_out -> rsqrt)
    float* norm_dst = ws + (size_t)N;                // N   (deg_in  -> rsqrt)
    float* hbuf     = ws + 2 * (size_t)N;            // 64N scaled features
    float* agg      = hbuf + 64 * (size_t)N;         // 64N scatter target
    float* hout     = agg + 64 * (size_t)N;          // 64N layer output (reused)
    float* sums     = hout + 64 * (size_t)N;         // G*64
    float* counts   = sums + (size_t)G * HID;        // G
    float* rs       = counts + G;                    // G

    auto nb = [](int n) { return (n + 255) / 256; };

    // degrees + norms
    k_zero<<<nb(2 * N), 256, 0, stream>>>(norm_src, 2 * N);
    k_zero<<<nb(G * HID + 2 * G), 256, 0, stream>>>(sums, G * HID + 2 * G);
    k_degree<<<nb(E), 256, 0, stream>>>(src, dst, norm_src, norm_dst, E);
    k_norm<<<nb(2 * N), 256, 0, stream>>>(norm_src, 2 * N);

    dim3 gGemm(N / 16, HID / 16);

    // ----- layer 1: K = 32 -----
    k_scale<<<nb(N * 32), 256, 0, stream>>>(feats, norm_src, hbuf, N, 5);
    k_zero<<<nb(N * 32), 256, 0, stream>>>(agg, N * 32);
    k_scatter<<<nb(E * 8), 256, 0, stream>>>(src, dst, hbuf, agg, E, 3, 32);
    k_gemm_wmma<<<gGemm, 32, 0, stream>>>(agg, norm_dst, W1, hout, 32, HID, 1);

    // ----- layer 2: K = 64 -----
    k_scale<<<nb(N * 64), 256, 0, stream>>>(hout, norm_src, hbuf, N, 6);
    k_zero<<<nb(N * 64), 256, 0, stream>>>(agg, N * 64);
    k_scatter<<<nb(E * 16), 256, 0, stream>>>(src, dst, hbuf, agg, E, 4, 64);
    k_gemm_wmma<<<gGemm, 32, 0, stream>>>(agg, norm_dst, W2, hout, 64, HID, 1);

    // ----- readout: segment mean + classifier -----
    int nblocks = 512;
    int chunk = (N + nblocks - 1) / nblocks;
    k_graphsum<<<nblocks, 64, 0, stream>>>(hout, gid, sums, counts, N, chunk);
    k_rscale<<<1, G, 0, stream>>>(counts, rs);
    dim3 gCls(G / 16, 1);   // [64,64] @ [64,16]
    k_gemm_wmma<<<gCls, 32, 0, stream>>>(sums, rs, Wc, out, HID, 16, 0);
}